// MxQueryAttentionMasked_38190849196235
// MI455X (gfx1250) — compile-verified
//
#include <hip/hip_runtime.h>
#include <math.h>

typedef float v2f __attribute__((ext_vector_type(2)));
typedef float v8f __attribute__((ext_vector_type(8)));
typedef unsigned int u32x4 __attribute__((ext_vector_type(4)));
typedef int i32x8 __attribute__((ext_vector_type(8)));
typedef int i32x4 __attribute__((ext_vector_type(4)));

#define B_DIM 4096
#define L_DIM 200
#define D_DIM 128
#define H_DIM 100
#define LP    208   // 13 * 16 (L padded)
#define HP    112   // 7 * 16  (H padded)
#define MT    13
#define NT    7
#define XSTR  132   // padded LDS row stride (floats) -> bank-conflict-free
#define NTHREADS 512
#define NWAVES   16
#define EPS_  1e-7f

// branch-free tanh: 1 - 2/(e^{2x}+1); saturates correctly at +/-inf.
__device__ __forceinline__ float tanh_fast(float x) {
  float e = __expf(2.0f * x);                       // v_exp_f32 path
  return 1.0f - 2.0f * __builtin_amdgcn_rcpf(e + 1.0f);
}

__global__ __launch_bounds__(NTHREADS) void mxqattn_kernel(
    const float* __restrict__ x, const float* __restrict__ q,
    const float* __restrict__ Wk, const float* __restrict__ bk,
    const float* __restrict__ Wq, const float* __restrict__ v,
    float* __restrict__ out)
{
  extern __shared__ float smem[];
  float* xs   = smem;                 // [LP][XSTR]  x tile (fp32), filled by TDM
  float* wkt  = xs  + LP * XSTR;      // [HP][XSTR]  Wk transposed [h][d]
  float* qb   = wkt + HP * XSTR;      // [HP]        bk + q@Wq   (bias per h)
  float* vv   = qb  + HP;             // [HP]        v padded
  float* sc   = vv  + HP;             // [LP]        scores (one owner wave per row)
  float* wgt  = sc  + LP;             // [LP]        weights
  float* msk  = wgt + LP;             // [LP]        row mask
  float* red  = msk + LP;             // [NWAVES]    block reduction scratch
  float* part = red + NWAVES;         // [NTHREADS]  output partials

  const int b    = blockIdx.x;
  const int tid  = threadIdx.x;
  const int lane = tid & 31;
  const int wave = tid >> 5;

  // ---------------- init ----------------
  for (int i = tid; i < LP; i += NTHREADS) msk[i] = 0.f;
  for (int i = tid; i < (LP - L_DIM) * XSTR; i += NTHREADS)           // zero padded x rows
    xs[(L_DIM + i / XSTR) * XSTR + (i % XSTR)] = 0.f;
  for (int i = tid; i < (HP - H_DIM) * XSTR; i += NTHREADS)           // zero padded Wk^T rows
    wkt[(H_DIM + i / XSTR) * XSTR + (i % XSTR)] = 0.f;
  __syncthreads();

  // ---- TDM: one async DMA of x[b] (200x128 f32) -> LDS, hardware row padding ----
  // pad_interval=6 -> pad every 2^(6+1)=128 DWORDs; pad_amount=3 -> 4 DWORDs.
  // Net LDS row stride = 128+4 = 132 floats = XSTR.
  if (wave == 0) {
    const unsigned lds_base = (unsigned)(uintptr_t)xs;  // flat LDS addr[31:0] = LDS offset
    const unsigned long long ga =
        (unsigned long long)(uintptr_t)(x + (size_t)b * (L_DIM * D_DIM));
    u32x4 g0;
    g0[0] = 1u;                                        // count=1 (valid), user mode
    g0[1] = lds_base;                                  // lds_addr (bytes)
    g0[2] = (unsigned)(ga & 0xFFFFFFFFu);              // global_addr[31:0]
    g0[3] = (unsigned)((ga >> 32) & 0x01FFFFFFu)       // global_addr[56:32]
          | (2u << 30);                                // type = 2 ("image")
    i32x8 g1;
    g1[0] = (2 << 16)                                  // data_size = 4 bytes
          | (1 << 20)                                  // pad_enable
          | (6 << 22)                                  // pad_interval = 128 DWORDs
          | (3 << 25);                                 // pad_amount   = 4 DWORDs
    g1[1] = (D_DIM & 0xFFFF) << 16;                    // tensor_dim0 = 128  (bits 79:48)
    g1[2] = (L_DIM & 0xFFFF) << 16;                    // dim0_hi=0 | tensor_dim1 = 200
    g1[3] = (D_DIM << 16);                             // dim1_hi=0 | tile_dim0 = 128
    g1[4] = L_DIM;                                     // tile_dim1 = 200 | tile_dim2 = 0
    g1[5] = D_DIM;                                     // tensor_dim0_stride = 128
    g1[6] = 0;                                         // dim0_stride hi | dim1_stride lo
    g1[7] = 0;                                         // tensor_dim1_stride (unused, 2D)
    const i32x4 gz4 = {0, 0, 0, 0};
    const i32x8 gz8 = {0, 0, 0, 0, 0, 0, 0, 0};
    __builtin_amdgcn_tensor_load_to_lds(g0, g1, gz4, gz4, gz8, 0);
  }

  // ---- overlap with the DMA: stage Wk transposed (L2-resident source) ----
  for (int i = tid; i < D_DIM * H_DIM; i += NTHREADS) {
    int d = i / H_DIM, h = i % H_DIM;
    wkt[h * XSTR + d] = Wk[i];
  }
  // ---- bias: qb[h] = q[b]·Wq[:,h] + bk[h]; vv[h] = v[h] ----
  if (tid < HP) {
    if (tid < H_DIM) {
      const float* qrow = q + (size_t)b * D_DIM;
      float acc = 0.f;
      for (int d = 0; d < D_DIM; ++d) acc += qrow[d] * Wq[d * H_DIM + tid];
      qb[tid] = acc + bk[tid];
      vv[tid] = v[tid];
    } else { qb[tid] = 0.f; vv[tid] = 0.f; }
  }

  if (wave == 0) __builtin_amdgcn_s_wait_tensorcnt(0); // DMA complete (issuing wave)
  __syncthreads();                                     // publish LDS to all waves

  // ---- row mask from the staged tile: any(x[l,:] != 0) ----
  for (int i = tid; i < (L_DIM * D_DIM) / 4; i += NTHREADS) {
    int row = i >> 5;                                  // 32 float4 per row
    int c4  = i & 31;
    const float4 vx = *(const float4*)(xs + row * XSTR + c4 * 4);
    if (vx.x != 0.f || vx.y != 0.f || vx.z != 0.f || vx.w != 0.f)
      msk[row] = 1.f;                                  // benign race: same value
  }
  __syncthreads();

  // ------- fp32 WMMA GEMM: wave owns one 16-row M block, loops all 7 N tiles -------
  // A (16x4 f32): lane = M row; K{0,1} in regs for lanes 0-15, K{2,3} for lanes 16-31.
  // A fragments are N-invariant: preload all 32 K-steps (64 VGPRs) once.
  if (wave < MT) {                                     // wave-uniform: EXEC all-ones
    const int mt  = wave;
    const int row = mt * 16 + (lane & 15);
    const int kh  = (lane >> 4) * 2;
    const float* ap = xs + row * XSTR + kh;

    v2f areg[D_DIM / 4];
    #pragma unroll
    for (int kk = 0; kk < D_DIM / 4; ++kk)
      areg[kk] = *(const v2f*)(ap + kk * 4);

    float accj[8];
    #pragma unroll
    for (int j = 0; j < 8; ++j) accj[j] = 0.f;

    for (int nt = 0; nt < NT; ++nt) {
      const int n = nt * 16 + (lane & 15);             // column (H)
      const float bias = qb[n];
      const float vn   = vv[n];
      const float* bp  = wkt + n * XSTR + kh;
      v8f c0, c1;                                      // two independent WMMA chains
      #pragma unroll
      for (int j = 0; j < 8; ++j) { c0[j] = bias; c1[j] = 0.f; }
      #pragma unroll
      for (int kk = 0; kk < D_DIM / 4; kk += 2) {      // 32 x v_wmma_f32_16x16x4_f32
        v2f b0 = *(const v2f*)(bp + kk * 4);
        v2f b1 = *(const v2f*)(bp + kk * 4 + 4);
        c0 = __builtin_amdgcn_wmma_f32_16x16x4_f32(false, areg[kk],     false, b0, (short)0, c0, false, false);
        c1 = __builtin_amdgcn_wmma_f32_16x16x4_f32(false, areg[kk + 1], false, b1, (short)0, c1, false, false);
      }
      #pragma unroll
      for (int j = 0; j < 8; ++j)
        accj[j] += tanh_fast(c0[j] + c1[j]) * vn;      // per-lane partial over its columns
    }
    // one reduction per wave: sum each accj over the 16 lanes of its half
    #pragma unroll
    for (int j = 0; j < 8; ++j) {
      float val = accj[j];
      #pragma unroll
      for (int off = 1; off < 16; off <<= 1)
        val += __shfl_xor(val, off, 32);
      if ((lane & 15) == 0)                            // lanes 0 & 16: rows j and j+8
        sc[mt * 16 + j + ((lane >> 4) << 3)] = val;    // plain store: wave owns these rows
    }
  }
  __syncthreads();

  // ------- masked exp-normalize (faithful: plain exp, +EPS) -------
  float a_loc = 0.f;
  if (tid < L_DIM) { a_loc = __expf(sc[tid]) * msk[tid]; wgt[tid] = a_loc; }
  float s = a_loc;
  #pragma unroll
  for (int off = 16; off >= 1; off >>= 1) s += __shfl_xor(s, off, 32);
  if (lane == 0) red[wave] = s;
  __syncthreads();
  if (wave == 0) {
    float tsum = (lane < NWAVES) ? red[lane] : 0.f;
    #pragma unroll
    for (int off = 16; off >= 1; off >>= 1) tsum += __shfl_xor(tsum, off, 32);
    if (lane == 0) red[0] = tsum + EPS_;
  }
  __syncthreads();
  const float inv = 1.f / red[0];
  if (tid < L_DIM) wgt[tid] *= inv;
  __syncthreads();

  // ------- out[b,d] = sum_l x[l,d] * w[l]  (x reused from LDS) -------
  {
    const int d  = tid & (D_DIM - 1);
    const int ps = tid >> 7;                           // 4 slices of 50 rows
    float acc = 0.f;
    for (int l = ps * 50; l < ps * 50 + 50; ++l)
      acc += xs[l * XSTR + d] * wgt[l];
    part[tid] = acc;
  }
  __syncthreads();
  if (tid < D_DIM)
    out[(size_t)b * D_DIM + tid] =
        part[tid] + part[tid + 128] + part[tid + 256] + part[tid + 384];
}

extern "C" void kernel_launch(void* const* d_in, const int* in_sizes, int n_in,
                              void* d_out, int out_size, void* d_ws, size_t ws_size,
                              hipStream_t stream) {
  const float* x  = (const float*)d_in[0];
  const float* q  = (const float*)d_in[1];
  const float* Wk = (const float*)d_in[2];
  const float* bk = (const float*)d_in[3];
  const float* Wq = (const float*)d_in[4];
  const float* v  = (const float*)d_in[5];
  float* out = (float*)d_out;

  const int B = in_sizes[1] / D_DIM;   // q is [B, D]
  const size_t shmem =
      (size_t)(LP * XSTR + HP * XSTR + HP + HP + LP + LP + LP + NWAVES + NTHREADS)
      * sizeof(float);                 // ~174.5 KB (fits 320 KB WGP LDS)

  // Allow >64KB dynamic LDS (idempotent, not a stream op — safe under capture).
  (void)hipFuncSetAttribute((const void*)mxqattn_kernel,
                            hipFuncAttributeMaxDynamicSharedMemorySize, (int)shmem);

  mxqattn_kernel<<<B, NTHREADS, shmem, stream>>>(x, q, Wk, bk, Wq, v, out);
}